// ExtractSplitPosition_73607149519526
// MI455X (gfx1250) — compile-verified
//
#include <hip/hip_runtime.h>
#include <stdint.h>

// ---------------------------------------------------------------------------
// Greedy 1-D NMS over B rows of F anchors (reference: ExtractSplitPosition).
//   scores  = sigmoid(logits)                 [B,F]
//   pos     = delta*16 + (i+0.5)*16           [B,F,2]
//   50x: argmax over masked scores, emit, suppress |mean - mean[pick]| <= 16
//
// MI455X mapping:
//   * per-row staging global->LDS via CDNA5 async-LDS ops (ASYNCcnt path)
//   * packed-u64-key wave32 shuffle argmax (score bits | ~index), registers only
//   * one 256-thread block (8 waves) per row, persistent grid, 2 barriers/pick
// ---------------------------------------------------------------------------

#define MAXF        2048
#define BLOCK       256
#define PT          (MAXF / BLOCK)     // 8 entries per thread
#define NWAVES      (BLOCK / 32)
#define MAX_OUT     50
#define FEAT_STRIDE 16.0f
#define SCORE_TH    0.7f
#define DIST_TH     16.0f

typedef unsigned long long u64;

// Generic pointer into LDS aperture: low 32 bits are the LDS byte offset
// (ISA 10.2: relative to the wave's LDS base).
__device__ __forceinline__ uint32_t lds_off32(const void* p) {
    return (uint32_t)(uintptr_t)p;
}

// CDNA5 async global->LDS copy, 16 bytes per lane. Tracked by ASYNCcnt.
__device__ __forceinline__ void async_ld_b128(uint32_t lds, const float* g) {
    asm volatile("global_load_async_to_lds_b128 %0, %1, off"
                 :: "v"(lds), "v"(g)
                 : "memory");
}

__device__ __forceinline__ void wait_async0() {
    asm volatile("s_wait_asynccnt 0" ::: "memory");
}

__global__ __launch_bounds__(BLOCK) void nms_split_kernel(
    const float* __restrict__ logits,   // [B,F]
    const float* __restrict__ deltas,   // [B,F,2]
    const int*   __restrict__ img_w_p,  // scalar
    float*       __restrict__ out,      // [B,50,2] positions ++ [B,50] scores
    int n_logit)                        // B*F
{
    __shared__ __align__(16) float  s_logit[MAXF];      //  8 KB staged
    __shared__ __align__(16) float2 s_delta[MAXF];      // 16 KB staged
    __shared__ __align__(16) float2 s_pos[MAXF];        // 16 KB computed
    __shared__ __align__(16) float  s_mean[MAXF];       //  8 KB computed
    __shared__ u64 s_rk[NWAVES];                        // wave-leader keys
    __shared__ u64 s_pk;                                // broadcast picked key

    const int t    = (int)threadIdx.x;
    const int lane = t & 31;
    const int wv   = t >> 5;

    const int   img_w = *img_w_p;
    const int   F     = img_w / 16;      // FEAT_STRIDE
    const int   B     = n_logit / F;
    const float w1    = (float)img_w - 1.0f;

    float* out_pos = out;                                   // [B,50,2]
    float* out_sc  = out + (size_t)B * MAX_OUT * 2;         // [B,50]

    for (int b = (int)blockIdx.x; b < B; b += (int)gridDim.x) {
        // ---- stage this row into LDS via async copies (16B chunks) ----
        const float* lrow = logits + (size_t)b * F;
        const float* drow = deltas + (size_t)b * F * 2;
        const int nlc = F >> 2;   // 16B chunks of logits
        const int ndc = F >> 1;   // 16B chunks of deltas (2F floats)
        for (int c = t; c < nlc; c += BLOCK)
            async_ld_b128(lds_off32(&s_logit[c * 4]), lrow + c * 4);
        for (int c = t; c < ndc; c += BLOCK)
            async_ld_b128(lds_off32(&s_delta[c * 2]), drow + c * 4);
        wait_async0();
        __syncthreads();

        // ---- compute orderable score keys / means (registers + LDS) ----
        // key = float_bits(sigmoid) ^ 0x80000000 for valid entries (all > 0
        // since score >= 0.7), 0 for invalid or suppressed. Low word of the
        // packed u64 key is ~index so ties pick the lowest index (jnp.argmax).
        unsigned kk[PT];
        float    mn[PT];
        #pragma unroll
        for (int k = 0; k < PT; ++k) {
            const int i = k * BLOCK + t;
            if (i < F) {
                const float  lg = s_logit[i];
                const float2 d  = s_delta[i];
                const float  c  = ((float)i + 0.5f) * FEAT_STRIDE;
                const float  p0 = fmaf(d.x, FEAT_STRIDE, c);
                const float  p1 = fmaf(d.y, FEAT_STRIDE, c);
                const float  s  = 1.0f / (1.0f + __expf(-lg));
                const bool valid = (p0 >= 0.0f) & (p0 <= w1) &
                                   (p1 >= 0.0f) & (p1 <= w1) &
                                   (s >= SCORE_TH);
                kk[k] = valid ? (__float_as_uint(s) ^ 0x80000000u) : 0u;
                mn[k] = 0.5f * (p0 + p1);
                s_mean[i] = mn[k];
                s_pos[i]  = make_float2(p0, p1);
            } else {
                kk[k] = 0u;
                mn[k] = 3.0e9f;   // never within DIST_TH of a real mean
            }
        }
        __syncthreads();

        // ---- greedy NMS: 50 sequential block-wide argmax picks ----
        for (int j = 0; j < MAX_OUT; ++j) {
            // local max over this thread's register slice (packed keys)
            u64 K = ((u64)kk[0] << 32) | (unsigned)~(unsigned)t;
            #pragma unroll
            for (int k = 1; k < PT; ++k) {
                const unsigned i = (unsigned)(k * BLOCK + t);
                const u64 C = ((u64)kk[k] << 32) | (unsigned)~i;
                if (C > K) K = C;
            }
            // wave32 xor-shuffle max (one u64 compare per step)
            #pragma unroll
            for (int off = 16; off >= 1; off >>= 1) {
                const u64 oK = __shfl_xor(K, off, 32);
                if (oK > K) K = oK;
            }
            if (lane == 0) s_rk[wv] = K;
            __syncthreads();

            if (wv == 0) {
                u64 K8 = (lane < NWAVES) ? s_rk[lane] : 0ull;
                #pragma unroll
                for (int off = 4; off >= 1; off >>= 1) {
                    const u64 oK = __shfl_xor(K8, off, 32);
                    if (oK > K8) K8 = oK;
                }
                if (lane == 0) {
                    s_pk = K8;
                    const unsigned kv  = (unsigned)(K8 >> 32);
                    const unsigned idx = ~(unsigned)K8;
                    const bool found = (kv != 0u);
                    const float2 pp = found ? s_pos[idx] : make_float2(0.0f, 0.0f);
                    const float  sv = found ? __uint_as_float(kv ^ 0x80000000u) : 0.0f;
                    const size_t o  = (size_t)b * MAX_OUT + j;
                    out_pos[o * 2 + 0] = pp.x;
                    out_pos[o * 2 + 1] = pp.y;
                    out_sc[o]          = sv;
                }
            }
            __syncthreads();

            // broadcast pick, then suppress on the register slice
            const u64      Kp  = s_pk;
            const unsigned kv  = (unsigned)(Kp >> 32);
            const bool     found = (kv != 0u);
            const unsigned idx = ~(unsigned)Kp;
            const float    pm  = s_mean[found ? idx : 0u];
            #pragma unroll
            for (int k = 0; k < PT; ++k)
                if (found && fabsf(mn[k] - pm) <= DIST_TH) kk[k] = 0u;
            // no third barrier: s_rk was last read before the 2nd barrier; next s_pk write happens only after
            // every thread passes the next iteration's first barrier
        }
    }
}

extern "C" void kernel_launch(void* const* d_in, const int* in_sizes, int n_in,
                              void* d_out, int out_size, void* d_ws, size_t ws_size,
                              hipStream_t stream) {
    const float* logits = (const float*)d_in[0];   // [B,F] f32
    const float* deltas = (const float*)d_in[1];   // [B,F,2] f32
    const int*   img_w  = (const int*)d_in[2];     // scalar int
    float*       out    = (float*)d_out;           // [B,50,2] ++ [B,50] f32
    const int n_logit   = in_sizes[0];

    dim3 grid(4096), block(BLOCK);
    hipLaunchKernelGGL(nms_split_kernel, grid, block, 0, stream,
                       logits, deltas, img_w, out, n_logit);
}